// Pointer_44332652429983
// MI455X (gfx1250) — compile-verified
//
#include <hip/hip_runtime.h>
#include <stdint.h>

typedef __attribute__((ext_vector_type(2))) float v2f;
typedef __attribute__((ext_vector_type(8))) float v8f;

#define B_SZ   32
#define L_SZ   4096
#define D_INP  150
#define D_ST   75
#define HID    75
#define KDIM   225   // D_ST + D_INP
#define KPAD   228   // next multiple of 4
#define NPAD   80    // 5 tiles of 16 (HID=75 padded)
#define NT     5
#define KCHUNK 57    // KPAD / 4
#define NPAIR  114   // KPAD / 2 (k-pairs)
#define INF_F  1.0e30f

// ---------------------------------------------------------------------------
// Kernel 1: per-row logits  s1[b,l] = tanh(u @ W0 + b0) @ W1 + b1 - INF*(1-mask)
// u = [state[b] (75) | inputs[b,l] (150)], padded K 225->228, N 75->80.
// One wave = one 16-row M tile, all 5 N tiles; fp32 WMMA 16x16x4.
// W0 staged into LDS with async global->LDS loads, pair-swizzled so each
// B fragment is a single aligned ds_load_b64 into an even VGPR pair.
// LDS layout: W0p[ ((k>>1)*NPAD + n)*2 + (k&1) ]
// ---------------------------------------------------------------------------
__global__ __launch_bounds__(256) void logits_kernel(
    const float* __restrict__ inputs,
    const float* __restrict__ state,
    const unsigned char* __restrict__ c_mask,
    const float* __restrict__ W0,
    const float* __restrict__ b0,
    const float* __restrict__ W1,
    const float* __restrict__ b1,
    float* __restrict__ s1_out)
{
    __shared__ float W0p[NPAIR * NPAD * 2];   // 72,960 B, pair-swizzled
    __shared__ float stateS[NPAD];
    __shared__ float b0s[NPAD];
    __shared__ float W1s[NPAD];

    const int tid = threadIdx.x;
    const int blk = blockIdx.x;
    const int b   = blk / (L_SZ / 128);          // 128 rows per block, same batch
    const int l0  = (blk % (L_SZ / 128)) * 128;

    // ---- async-stage W0 into LDS with pair swizzle (ASYNCcnt path) ----
    const uint32_t ldsBase = (uint32_t)(uintptr_t)(&W0p[0]); // low 32b = LDS addr
    for (int idx = tid; idx < KDIM * HID; idx += 256) {
        const int k = idx / HID;
        const int n = idx - k * HID;
        const uint32_t dst =
            ldsBase + (uint32_t)((((k >> 1) * NPAD + n) << 1) + (k & 1)) * 4u;
        const uint64_t src = (uint64_t)(uintptr_t)(W0 + idx);  // row-major [k][n]
        asm volatile("global_load_async_to_lds_b32 %0, %1, off"
                     :: "v"(dst), "v"(src) : "memory");
    }
    // ---- zero the pad cells (disjoint from async destinations) ----
    for (int idx = tid; idx < NPAIR * NPAD * 2; idx += 256) {
        const int v  = idx & 1;
        const int pn = idx >> 1;
        const int n  = pn % NPAD;
        const int k  = (pn / NPAD) * 2 + v;
        if (n >= HID || k >= KDIM) W0p[idx] = 0.0f;
    }
    if (tid < NPAD) {
        stateS[tid] = (tid < D_ST) ? state[b * D_ST + tid] : 0.0f;
        b0s[tid]    = (tid < HID)  ? b0[tid] : 0.0f;
        W1s[tid]    = (tid < HID)  ? W1[tid] : 0.0f;
    }
    asm volatile("s_wait_asynccnt 0x0" ::: "memory");
    __syncthreads();

    const int wave   = tid >> 5;
    const int lane   = tid & 31;
    const int half   = lane >> 4;   // 0: K pair base, 1: +2
    const int ln     = lane & 15;
    const int l_tile = l0 + wave * 16;
    const int m      = ln;          // A-matrix row owned by this lane

    const float* __restrict__ inrow =
        inputs + (size_t)(b * L_SZ + l_tile + m) * D_INP;
    __builtin_prefetch(inrow, 0, 3);

    v8f zero = {};
    v8f acc[NT];
#pragma unroll
    for (int nb = 0; nb < NT; ++nb) acc[nb] = zero;

    // K loop: 57 chunks of 4.  A frag: VGPR v = u[m][4*kc + 2*half + v].
    //                          B frag: VGPR v = W0[4*kc + 2*half + v][n].
#pragma unroll
    for (int kc = 0; kc < KCHUNK; ++kc) {
        const int c0 = kc * 4 + half * 2;     // always even
        v2f a;
        if (kc < 18) {                        // c0+1 <= 73 < 75: pure state
            a = *(const v2f*)&stateS[c0];     // aligned ds_load_b64
        } else if (kc == 18) {                // boundary chunk (74/75)
            a.x = (c0     < D_ST) ? stateS[c0]     : inrow[c0 - D_ST];
            a.y = (c0 + 1 < D_ST) ? stateS[c0 + 1] : inrow[c0 + 1 - D_ST];
        } else if (kc < KCHUNK - 1) {         // pure input region
            a.x = inrow[c0 - D_ST];
            a.y = inrow[c0 - D_ST + 1];
        } else {                              // kc == 56: K padding (225..227)
            a.x = (c0 < KDIM) ? inrow[c0 - D_ST] : 0.0f;
            a.y = 0.0f;
        }
        const int p = c0 >> 1;                // pair index = 2*kc + half
#pragma unroll
        for (int nb = 0; nb < NT; ++nb) {
            const v2f bf =
                *(const v2f*)&W0p[((p * NPAD + nb * 16 + ln) << 1)]; // b64, even pair
            acc[nb] = __builtin_amdgcn_wmma_f32_16x16x4_f32(
                false, a, false, bf, (short)0, acc[nb], false, false);
        }
    }

    // Epilogue: s0 = tanh(acc + b0), per-row s = sum_n s0*W1[n].
    // C layout: VGPR i = row (i + 8*half), lane%16 = column n within tile.
    float row_s[8];
#pragma unroll
    for (int i = 0; i < 8; ++i) row_s[i] = 0.0f;
#pragma unroll
    for (int nb = 0; nb < NT; ++nb) {
        const int n = nb * 16 + ln;
        const float bias = b0s[n];
        const float w1v  = W1s[n];
#pragma unroll
        for (int i = 0; i < 8; ++i)
            row_s[i] += tanhf(acc[nb][i] + bias) * w1v;
    }
    // Cross-lane sum within each 16-lane half (offsets 1..8 never cross halves).
#pragma unroll
    for (int i = 0; i < 8; ++i) {
        float v = row_s[i];
        v += __shfl_xor(v, 1, 32);
        v += __shfl_xor(v, 2, 32);
        v += __shfl_xor(v, 4, 32);
        v += __shfl_xor(v, 8, 32);
        row_s[i] = v;
    }
    if (ln == 0) {  // lane 0 -> rows 0..7, lane 16 -> rows 8..15
        const float bb1 = b1[0];
        const int lrow0 = l_tile + half * 8;
#pragma unroll
        for (int i = 0; i < 8; ++i) {
            const int l = lrow0 + i;
            const float sv = row_s[i] + bb1;
            const float mk = c_mask[b * L_SZ + l] ? 1.0f : 0.0f;
            s1_out[b * L_SZ + l] = sv - INF_F * (1.0f - mk);
        }
    }
}

// ---------------------------------------------------------------------------
// Kernel 2: per-batch softmax over L and weighted feature reduction.
// One block per batch; memory-bound, fully coalesced input reads.
// ---------------------------------------------------------------------------
__global__ __launch_bounds__(1024) void softmax_wsum_kernel(
    const float* __restrict__ inputs,
    const float* __restrict__ s1,
    float* __restrict__ res)
{
    __shared__ float wbuf[L_SZ];
    __shared__ float red[1024];

    const int b   = blockIdx.x;
    const int tid = threadIdx.x;
    const float* __restrict__ ss = s1 + (size_t)b * L_SZ;

    // ---- max over L ----
    float lmax = -3.402823466e38f;
    for (int l = tid; l < L_SZ; l += 1024) lmax = fmaxf(lmax, ss[l]);
    red[tid] = lmax;
    __syncthreads();
    for (int off = 512; off > 0; off >>= 1) {
        if (tid < off) red[tid] = fmaxf(red[tid], red[tid + off]);
        __syncthreads();
    }
    const float mx = red[0];
    __syncthreads();

    // ---- exp + sum ----
    float lsum = 0.0f;
    for (int l = tid; l < L_SZ; l += 1024) {
        const float w = __expf(ss[l] - mx);
        wbuf[l] = w;
        lsum += w;
    }
    red[tid] = lsum;
    __syncthreads();
    for (int off = 512; off > 0; off >>= 1) {
        if (tid < off) red[tid] += red[tid + off];
        __syncthreads();
    }
    const float inv_den = 1.0f / red[0];
    __syncthreads();

    // ---- weighted sum of inputs: res[b, d] = sum_l w[l] * inputs[b,l,d] ----
    const int tx = tid & 255;   // feature d (only 0..149 active)
    const int ty = tid >> 8;    // 0..3 : L slice
    float acc = 0.0f;
    if (tx < D_INP) {
        const float* __restrict__ inb = inputs + (size_t)b * L_SZ * D_INP;
#pragma unroll 8
        for (int l = ty; l < L_SZ; l += 4)
            acc += wbuf[l] * inb[(size_t)l * D_INP + tx];
    }
    red[tid] = acc;             // tid == ty*256 + tx
    __syncthreads();
    if (ty == 0 && tx < D_INP) {
        const float tot = red[tx] + red[256 + tx] + red[512 + tx] + red[768 + tx];
        res[b * D_INP + tx] = tot * inv_den;
    }
}

// ---------------------------------------------------------------------------
extern "C" void kernel_launch(void* const* d_in, const int* in_sizes, int n_in,
                              void* d_out, int out_size, void* d_ws, size_t ws_size,
                              hipStream_t stream)
{
    (void)in_sizes; (void)n_in; (void)out_size; (void)d_ws; (void)ws_size;

    const float*         inputs = (const float*)d_in[0];
    const float*         state  = (const float*)d_in[1];
    const unsigned char* c_mask = (const unsigned char*)d_in[2];
    const float*         W0     = (const float*)d_in[3];
    const float*         b0     = (const float*)d_in[4];
    const float*         W1     = (const float*)d_in[5];
    const float*         b1     = (const float*)d_in[6];

    float* out = (float*)d_out;
    float* res = out;                    // [B, 1, D_INP]   (4800 floats)
    float* s1  = out + B_SZ * D_INP;     // [B, L]          (131072 floats)

    logits_kernel<<<dim3((B_SZ * L_SZ) / 128), dim3(256), 0, stream>>>(
        inputs, state, c_mask, W0, b0, W1, b1, s1);

    softmax_wsum_kernel<<<dim3(B_SZ), dim3(1024), 0, stream>>>(
        inputs, s1, res);
}